// Sparsify_72258529788638
// MI455X (gfx1250) — compile-verified
//
#include <hip/hip_runtime.h>
#include <stdint.h>

// Block-top-K sparsify: out = x * mask, mask keeps the 4 largest scores per
// contiguous 8-float block (jnp stable-argsort tie-breaking: lower index
// pruned first on ties).
//
// Roofline: 768 MB single-touch traffic -> ~33 us at 23.3 TB/s; no matrix
// math, so no WMMA. CDNA5 paths used instead:
//   - score staged global->LDS with global_load_async_to_lds_b128 (ASYNCcnt),
//     double-buffered, drained with exact s_wait_asynccnt values.
//   - all three streams carry non-temporal hints (single touch; don't thrash
//     the 192 MB L2): th:TH_LOAD_NT on the async DMA, __builtin_nontemporal_*
//     for x loads / out stores.
//   - wave32 sizing: 256-thread WGs = 8 waves; 16 KB LDS of 320 KB/WGP.

typedef float v4f __attribute__((ext_vector_type(4)));

#define BLK 8      // block size along last dim
#define KEEP 4     // non-zeros kept per block
#define TPB 256    // 8 waves per workgroup (wave32)
#define GRID 4096

__global__ __launch_bounds__(TPB) void sparsify_blocktopk(
    const float* __restrict__ x, const float* __restrict__ score,
    float* __restrict__ out, int nBlocks, int numIters, unsigned gsize) {
  // double-buffered LDS staging for score: 2 * 256 threads * 32 B = 16 KB
  __shared__ __align__(16) unsigned char sbuf[2 * TPB * 32];

  const int tid = threadIdx.x;
  const unsigned gtid = blockIdx.x * blockDim.x + tid;

  // low 32 bits of a generic pointer into LDS == byte offset in LDS space
  const unsigned ldsBase =
      (unsigned)(uintptr_t)(void*)sbuf + (unsigned)tid * 32u;

  // Issue one 32B stage (two async b128 copies) for clamped block index `bn`
  // into buffer `buf`. Addresses are clamped (never predicated) so every wave
  // issues exactly 2 async ops per stage -> s_wait_asynccnt counts are exact.
  auto stage = [&](long long bn, int buf) {
    unsigned long long ga =
        (unsigned long long)(uintptr_t)score + (unsigned long long)bn * 32ull;
    unsigned la = ldsBase + (unsigned)buf * (TPB * 32u);
    asm volatile("global_load_async_to_lds_b128 %0, %1, off th:TH_LOAD_NT"
                 :: "v"(la), "v"(ga) : "memory");
    asm volatile("global_load_async_to_lds_b128 %0, %1, off th:TH_LOAD_NT"
                 :: "v"(la + 16u), "v"(ga + 16ull) : "memory");
  };

  long long b = (long long)gtid;
  {
    long long b0 = b < nBlocks ? b : (long long)nBlocks - 1;
    stage(b0, 0);
  }

  int buf = 0;
  for (int it = 0; it < numIters; ++it) {
    const bool p = b < nBlocks;
    const long long bc = p ? b : (long long)nBlocks - 1;  // clamped

    if (it + 1 < numIters) {  // uniform branch
      long long bn = b + (long long)gsize;
      if (bn >= nBlocks) bn = (long long)nBlocks - 1;
      stage(bn, buf ^ 1);
      // prefetch next iteration's x tile while the DMA runs
      __builtin_prefetch(x + (unsigned long long)bn * BLK, 0, 0);
      asm volatile("s_wait_asynccnt 2" ::: "memory");  // stage(it) complete
    } else {
      asm volatile("s_wait_asynccnt 0" ::: "memory");  // drain
    }

    // ---- consume staged scores from LDS (ds_load_b128 x2) ----
    const v4f* sp =
        (const v4f*)(sbuf + (size_t)buf * (TPB * 32) + (size_t)tid * 32);
    v4f sa = sp[0];
    v4f sb = sp[1];

    // ---- x: 128-bit non-temporal stream loads ----
    const v4f* xp = (const v4f*)(x + (unsigned long long)bc * BLK);
    v4f xa = __builtin_nontemporal_load(xp);
    v4f xb = __builtin_nontemporal_load(xp + 1);

    // ---- stable-sort rank via 28 pairwise compares ----
    // rank[j] = #elements preceding j in ascending stable sort.
    // For i<j: i precedes j iff s[i] <= s[j] (ties keep index order).
    float sc[BLK] = {sa.x, sa.y, sa.z, sa.w, sb.x, sb.y, sb.z, sb.w};
    float xv[BLK] = {xa.x, xa.y, xa.z, xa.w, xb.x, xb.y, xb.z, xb.w};
    int rk[BLK] = {0, 0, 0, 0, 0, 0, 0, 0};
#pragma unroll
    for (int i = 0; i < BLK; ++i) {
#pragma unroll
      for (int j = i + 1; j < BLK; ++j) {
        bool le = sc[i] <= sc[j];
        rk[j] += le ? 1 : 0;
        rk[i] += le ? 0 : 1;
      }
    }
    float ov[BLK];
#pragma unroll
    for (int j = 0; j < BLK; ++j)
      ov[j] = (rk[j] >= (BLK - KEEP)) ? xv[j] : 0.0f;  // keep top-4
    v4f oa = {ov[0], ov[1], ov[2], ov[3]};
    v4f ob = {ov[4], ov[5], ov[6], ov[7]};

    if (p) {
      v4f* op = (v4f*)(out + (unsigned long long)b * BLK);
      __builtin_nontemporal_store(oa, op);
      __builtin_nontemporal_store(ob, op + 1);
    }

    b += (long long)gsize;
    buf ^= 1;
  }
}

extern "C" void kernel_launch(void* const* d_in, const int* in_sizes, int n_in,
                              void* d_out, int out_size, void* d_ws,
                              size_t ws_size, hipStream_t stream) {
  (void)in_sizes; (void)n_in; (void)d_ws; (void)ws_size;
  const float* x = (const float*)d_in[0];
  const float* score = (const float*)d_in[1];
  float* out = (float*)d_out;
  int nBlocks = out_size / BLK;  // 8,388,608 for 8192x8192
  if (nBlocks <= 0) return;
  const unsigned gsize = (unsigned)GRID * TPB;           // 1,048,576 threads
  const int numIters = (int)(((unsigned)nBlocks + gsize - 1u) / gsize);  // 8
  sparsify_blocktopk<<<GRID, TPB, 0, stream>>>(x, score, out, nBlocks,
                                               numIters, gsize);
}